// VectorQuantizer_29583734734903
// MI455X (gfx1250) — compile-verified
//
#include <hip/hip_runtime.h>
#include <math.h>

typedef float v2f __attribute__((ext_vector_type(2)));
typedef float v8f __attribute__((ext_vector_type(8)));
typedef int   v4i __attribute__((ext_vector_type(4)));

#define N_TOK 262144
#define KC    256
#define DIM   32
#define ROWS  128
#define NBLK  (N_TOK / ROWS)   // 2048
#define XPAD  36               // 16B-aligned rows, conflict-free b64/b128 LDS access
#define BETA    0.25f
#define SK_EPS  0.01f
#define LOG2E   1.4426950408889634f
#define LOG2B   18.0f          // log2(262144)
#define LOG2K   8.0f           // log2(256)

// ws layout (float slots)
#define WS_DMIN   0
#define WS_DMAX   1
#define WS_MID    2
#define WS_SCALE2 3
#define WS_B2     8            // 256 floats
#define WS_LOSS   264          // 2048 floats
#define WS_PART   4096         // 2048*256*2 floats (m,s pairs)

#if __has_builtin(__builtin_amdgcn_global_load_async_to_lds_b128)
#define HAS_ASYNC_LDS 1
#else
#define HAS_ASYNC_LDS 0
#endif

__device__ __forceinline__ void lse_merge(float& m, float& s, float m2, float s2) {
  float M = fmaxf(m, m2);
  s = s * exp2f(m - M) + s2 * exp2f(m2 - M);
  m = M;
}

#if HAS_ASYNC_LDS
typedef __attribute__((address_space(1))) v4i* gv4i_p;
typedef __attribute__((address_space(3))) v4i* lv4i_p;
__device__ __forceinline__ void async_cp16(const float* g, float* l) {
  __builtin_amdgcn_global_load_async_to_lds_b128(
      (gv4i_p)const_cast<float*>(g), (lv4i_p)l, 0, 0);
}
__device__ __forceinline__ void async_wait0() {
#if __has_builtin(__builtin_amdgcn_s_wait_asynccnt)
  __builtin_amdgcn_s_wait_asynccnt(0);
#else
  asm volatile("s_wait_asynccnt 0x0" ::: "memory");
#endif
}
#endif

// Cooperative load of w (256x32) and x tile (128x32) into padded LDS.
__device__ __forceinline__ void load_tiles(const float* __restrict__ x,
                                           const float* __restrict__ w,
                                           float* w_lds, float* x_lds, int rowbase) {
  int t = threadIdx.x;
#if HAS_ASYNC_LDS
#pragma unroll
  for (int u = 0; u < 8; ++u) {            // 2048 x 16B of w
    int p = t + 256 * u; int r = p >> 3, q = p & 7;
    async_cp16(w + r * DIM + 4 * q, &w_lds[r * XPAD + 4 * q]);
  }
#pragma unroll
  for (int u = 0; u < 4; ++u) {            // 1024 x 16B of x tile
    int p = t + 256 * u; int r = p >> 3, q = p & 7;
    async_cp16(x + (size_t)(rowbase + r) * DIM + 4 * q, &x_lds[r * XPAD + 4 * q]);
  }
  async_wait0();
#else
#pragma unroll
  for (int u = 0; u < 8; ++u) {
    int p = t + 256 * u; int r = p >> 3, q = p & 7;
    float4 v = *(const float4*)(w + r * DIM + 4 * q);
    *(float4*)&w_lds[r * XPAD + 4 * q] = v;   // 16B-aligned (XPAD=36)
  }
#pragma unroll
  for (int u = 0; u < 4; ++u) {
    int p = t + 256 * u; int r = p >> 3, q = p & 7;
    float4 v = *(const float4*)(x + (size_t)(rowbase + r) * DIM + 4 * q);
    *(float4*)&x_lds[r * XPAD + 4 * q] = v;
  }
#endif
}

__device__ __forceinline__ void load_afrag(const float* x_lds, v2f a[8], int lane, int wv) {
  int mrow = wv * 16 + (lane & 15);
  int ksel = (lane >> 4) << 1;             // 0 or 2
#pragma unroll
  for (int kk = 0; kk < 8; ++kk)
    a[kk] = *(const v2f*)&x_lds[mrow * XPAD + 4 * kk + ksel];
}

__device__ __forceinline__ v8f tile_mm(const float* w_lds, const v2f a[8], int ct, int lane) {
  int jc = ct * 16 + (lane & 15);
  int ksel = (lane >> 4) << 1;
  v8f acc = {};
#pragma unroll
  for (int kk = 0; kk < 8; ++kk) {
    v2f b = *(const v2f*)&w_lds[jc * XPAD + 4 * kk + ksel];
    acc = __builtin_amdgcn_wmma_f32_16x16x4_f32(false, a[kk], false, b,
                                                (short)0, acc, false, false);
  }
  return acc;
}

// ---------------- init ----------------
__global__ void k_init(float* ws) {
  int t = threadIdx.x;
  if (t == 0) {
    ((int*)ws)[WS_DMIN] = 0x7f7fffff;   // +FLT_MAX bits (ordered-int trick, d >= 0)
    ((int*)ws)[WS_DMAX] = 0;
  }
  ws[WS_B2 + t] = 0.0f;
}

// ---------------- pass 1: global min/max of d ----------------
__global__ void __launch_bounds__(256) k_minmax(const float* __restrict__ x,
                                                const float* __restrict__ w,
                                                float* ws) {
  __shared__ float w_lds[KC * XPAD];
  __shared__ float x_lds[ROWS * XPAD];
  __shared__ float wsq_lds[KC];
  __shared__ float red[16];
  int lane = threadIdx.x & 31, wv = threadIdx.x >> 5;
  int rowbase = blockIdx.x * ROWS;

  load_tiles(x, w, w_lds, x_lds, rowbase);
  __syncthreads();
  { int j = threadIdx.x; float s = 0.f;
#pragma unroll
    for (int k = 0; k < DIM; ++k) { float v = w_lds[j * XPAD + k]; s += v * v; }
    wsq_lds[j] = s; }
  __syncthreads();

  v2f a[8]; load_afrag(x_lds, a, lane, wv);
  float xp = 0.f;
#pragma unroll
  for (int kk = 0; kk < 8; ++kk) xp += a[kk].x * a[kk].x + a[kk].y * a[kk].y;
  float xs = xp + __shfl_xor(xp, 16);
  float rowXs[8];
#pragma unroll
  for (int c = 0; c < 8; ++c) rowXs[c] = __shfl(xs, ((lane & 16) >> 1) + c);

  float dmin = 3.4e38f, dmax = 0.f;
#pragma unroll
  for (int ct = 0; ct < 16; ++ct) {
    float wsq = wsq_lds[ct * 16 + (lane & 15)];
    v8f acc = tile_mm(w_lds, a, ct, lane);
#pragma unroll
    for (int c = 0; c < 8; ++c) {
      float d = fmaxf(rowXs[c] + wsq - 2.0f * acc[c], 0.0f);
      dmin = fminf(dmin, d); dmax = fmaxf(dmax, d);
    }
  }
#pragma unroll
  for (int dd = 1; dd < 32; dd <<= 1) {
    dmin = fminf(dmin, __shfl_xor(dmin, dd));
    dmax = fmaxf(dmax, __shfl_xor(dmax, dd));
  }
  if (lane == 0) { red[wv * 2] = dmin; red[wv * 2 + 1] = dmax; }
  __syncthreads();
  if (threadIdx.x == 0) {
    float mn = red[0], mx = red[1];
#pragma unroll
    for (int i = 1; i < 8; ++i) { mn = fminf(mn, red[i * 2]); mx = fmaxf(mx, red[i * 2 + 1]); }
    atomicMin((int*)ws + WS_DMIN, __float_as_int(mn));
    atomicMax((int*)ws + WS_DMAX, __float_as_int(mx));
  }
}

// ---------------- scalar setup ----------------
__global__ void k_setup(float* ws) {
  float dmin = __int_as_float(((int*)ws)[WS_DMIN]);
  float dmax = __int_as_float(((int*)ws)[WS_DMAX]);
  float mid = 0.5f * (dmax + dmin);
  float amp = fmaxf(dmax - mid, 1e-5f);
  ws[WS_MID] = mid;
  ws[WS_SCALE2] = LOG2E / (amp * SK_EPS);
}

// ------------- Sinkhorn row pass + column partials -------------
__global__ void __launch_bounds__(256) k_sink(const float* __restrict__ x,
                                              const float* __restrict__ w,
                                              float* ws) {
  __shared__ float w_lds[KC * XPAD];
  __shared__ float x_lds[ROWS * XPAD];
  __shared__ float wsq_lds[KC];
  __shared__ float b_lds[KC];
  __shared__ float wpart[8 * KC * 2];
  int lane = threadIdx.x & 31, wv = threadIdx.x >> 5;
  int rowbase = blockIdx.x * ROWS;

  load_tiles(x, w, w_lds, x_lds, rowbase);
  b_lds[threadIdx.x] = ws[WS_B2 + threadIdx.x];
  __syncthreads();
  { int j = threadIdx.x; float s = 0.f;
#pragma unroll
    for (int k = 0; k < DIM; ++k) { float v = w_lds[j * XPAD + k]; s += v * v; }
    wsq_lds[j] = s; }
  __syncthreads();

  float scale2 = ws[WS_SCALE2], mid = ws[WS_MID];
  v2f a[8]; load_afrag(x_lds, a, lane, wv);
  float xp = 0.f;
#pragma unroll
  for (int kk = 0; kk < 8; ++kk) xp += a[kk].x * a[kk].x + a[kk].y * a[kk].y;
  float xs = xp + __shfl_xor(xp, 16);
  float rowC[8];
#pragma unroll
  for (int c = 0; c < 8; ++c)
    rowC[c] = (mid - __shfl(xs, ((lane & 16) >> 1) + c)) * scale2;

  float colC[16], b2v[16];
#pragma unroll
  for (int ct = 0; ct < 16; ++ct) {
    int j = ct * 16 + (lane & 15);
    colC[ct] = -scale2 * wsq_lds[j];
    b2v[ct] = b_lds[j];
  }

  // scores in registers: tt[ct][c] = s0 (base-2 scaled)
  float tt[16][8];
#pragma unroll
  for (int ct = 0; ct < 16; ++ct) {
    v8f acc = tile_mm(w_lds, a, ct, lane);
#pragma unroll
    for (int c = 0; c < 8; ++c)
      tt[ct][c] = 2.0f * scale2 * acc[c] + rowC[c] + colC[ct];
  }

  // row LSE -> a2[c]
  float a2[8];
#pragma unroll
  for (int c = 0; c < 8; ++c) {
    float m = -1e30f;
#pragma unroll
    for (int ct = 0; ct < 16; ++ct) m = fmaxf(m, tt[ct][c] + b2v[ct]);
    float s = 0.f;
#pragma unroll
    for (int ct = 0; ct < 16; ++ct) s += exp2f(tt[ct][c] + b2v[ct] - m);
#pragma unroll
    for (int dd = 1; dd < 16; dd <<= 1)
      lse_merge(m, s, __shfl_xor(m, dd), __shfl_xor(s, dd));
    a2[c] = -LOG2B - m - log2f(s);
  }

  // column partials over this wave's 16 rows
#pragma unroll
  for (int ct = 0; ct < 16; ++ct) {
    float m = -1e30f;
#pragma unroll
    for (int c = 0; c < 8; ++c) m = fmaxf(m, tt[ct][c] + a2[c]);
    float s = 0.f;
#pragma unroll
    for (int c = 0; c < 8; ++c) s += exp2f(tt[ct][c] + a2[c] - m);
    lse_merge(m, s, __shfl_xor(m, 16), __shfl_xor(s, 16));
    if (lane < 16) {
      int j = ct * 16 + lane;
      wpart[(wv * KC + j) * 2]     = m;
      wpart[(wv * KC + j) * 2 + 1] = s;
    }
  }
  __syncthreads();

  // combine 8 wave partials -> per-block (m,s) to global ws
  { int j = threadIdx.x;
    float m = -1e30f, s = 0.f;
#pragma unroll
    for (int v = 0; v < 8; ++v)
      lse_merge(m, s, wpart[(v * KC + j) * 2], wpart[(v * KC + j) * 2 + 1]);
    size_t o = WS_PART + ((size_t)blockIdx.x * KC + j) * 2;
    ws[o] = m; ws[o + 1] = s;
  }
}

// ------------- column reduce -> new b2 -------------
__global__ void __launch_bounds__(256) k_colreduce(float* ws) {
  __shared__ float rm[256], rs[256];
  int j = blockIdx.x, t = threadIdx.x;
  float m = -1e30f, s = 0.f;
#pragma unroll
  for (int u = 0; u < 8; ++u) {
    int b = t + 256 * u;
    size_t o = WS_PART + ((size_t)b * KC + j) * 2;
    lse_merge(m, s, ws[o], ws[o + 1]);
  }
  rm[t] = m; rs[t] = s;
  __syncthreads();
  for (int off = 128; off > 0; off >>= 1) {
    if (t < off) {
      float mm = rm[t], ss = rs[t];
      lse_merge(mm, ss, rm[t + off], rs[t + off]);
      rm[t] = mm; rs[t] = ss;
    }
    __syncthreads();
  }
  if (t == 0) ws[WS_B2 + j] = -LOG2K - (rm[0] + log2f(fmaxf(rs[0], 1e-37f)));
}

// ------------- final: argmax, gather, outputs, loss partial -------------
__global__ void __launch_bounds__(256) k_final(const float* __restrict__ x,
                                               const float* __restrict__ w,
                                               float* __restrict__ out,
                                               float* ws) {
  __shared__ float w_lds[KC * XPAD];
  __shared__ float x_lds[ROWS * XPAD];
  __shared__ float wsq_lds[KC];
  __shared__ float b_lds[KC];
  __shared__ int   idx_lds[ROWS];
  __shared__ float lred[8];
  int lane = threadIdx.x & 31, wv = threadIdx.x >> 5;
  int rowbase = blockIdx.x * ROWS;

  load_tiles(x, w, w_lds, x_lds, rowbase);
  b_lds[threadIdx.x] = ws[WS_B2 + threadIdx.x];
  __syncthreads();
  { int j = threadIdx.x; float s = 0.f;
#pragma unroll
    for (int k = 0; k < DIM; ++k) { float v = w_lds[j * XPAD + k]; s += v * v; }
    wsq_lds[j] = s; }
  __syncthreads();

  float scale2 = ws[WS_SCALE2], mid = ws[WS_MID];
  v2f a[8]; load_afrag(x_lds, a, lane, wv);
  float xp = 0.f;
#pragma unroll
  for (int kk = 0; kk < 8; ++kk) xp += a[kk].x * a[kk].x + a[kk].y * a[kk].y;
  float xs = xp + __shfl_xor(xp, 16);
  float rowC[8];
#pragma unroll
  for (int c = 0; c < 8; ++c)
    rowC[c] = (mid - __shfl(xs, ((lane & 16) >> 1) + c)) * scale2;

  float bestv[8]; int besti[8];
#pragma unroll
  for (int c = 0; c < 8; ++c) { bestv[c] = -3.4e38f; besti[c] = 0; }

#pragma unroll
  for (int ct = 0; ct < 16; ++ct) {
    int j = ct * 16 + (lane & 15);
    float cc = -scale2 * wsq_lds[j] + b_lds[j];
    v8f acc = tile_mm(w_lds, a, ct, lane);
#pragma unroll
    for (int c = 0; c < 8; ++c) {
      float v = 2.0f * scale2 * acc[c] + rowC[c] + cc;
      if (v > bestv[c]) { bestv[c] = v; besti[c] = j; }
    }
  }
#pragma unroll
  for (int c = 0; c < 8; ++c) {
#pragma unroll
    for (int dd = 1; dd < 16; dd <<= 1) {
      float ov = __shfl_xor(bestv[c], dd);
      int   oi = __shfl_xor(besti[c], dd);
      if (ov > bestv[c] || (ov == bestv[c] && oi < besti[c])) { bestv[c] = ov; besti[c] = oi; }
    }
    if ((lane & 15) == 0)
      idx_lds[wv * 16 + ((lane & 16) >> 1) + c] = besti[c];
  }
  __syncthreads();

  // write x_q (coalesced float4), indices, accumulate loss
  float lacc = 0.f;
#pragma unroll
  for (int u = 0; u < 4; ++u) {
    int p = threadIdx.x + 256 * u; int r = p >> 3, q = p & 7;
    int id = idx_lds[r];
    float4 o;
    float d0 = w_lds[id * XPAD + 4 * q + 0], e0 = d0 - x_lds[r * XPAD + 4 * q + 0];
    float d1 = w_lds[id * XPAD + 4 * q + 1], e1 = d1 - x_lds[r * XPAD + 4 * q + 1];
    float d2 = w_lds[id * XPAD + 4 * q + 2], e2 = d2 - x_lds[r * XPAD + 4 * q + 2];
    float d3 = w_lds[id * XPAD + 4 * q + 3], e3 = d3 - x_lds[r * XPAD + 4 * q + 3];
    o.x = d0; o.y = d1; o.z = d2; o.w = d3;
    lacc += e0 * e0 + e1 * e1 + e2 * e2 + e3 * e3;
    *(float4*)(out + (size_t)(rowbase + r) * DIM + 4 * q) = o;
  }
  if (threadIdx.x < ROWS)
    out[(size_t)N_TOK * DIM + 1 + (size_t)rowbase + threadIdx.x] = (float)idx_lds[threadIdx.x];

#pragma unroll
  for (int dd = 1; dd < 32; dd <<= 1) lacc += __shfl_xor(lacc, dd);
  if (lane == 0) lred[wv] = lacc;
  __syncthreads();
  if (threadIdx.x == 0) {
    float s = 0.f;
#pragma unroll
    for (int i = 0; i < 8; ++i) s += lred[i];
    ws[WS_LOSS + blockIdx.x] = s;   // deterministic per-block partial
  }
}

// ------------- loss finalize (deterministic tree) -------------
__global__ void k_loss(float* __restrict__ out, const float* __restrict__ ws) {
  __shared__ float red[256];
  int t = threadIdx.x;
  float s = 0.f;
#pragma unroll
  for (int u = 0; u < 8; ++u) s += ws[WS_LOSS + t + 256 * u];
  red[t] = s;
  __syncthreads();
  for (int off = 128; off > 0; off >>= 1) {
    if (t < off) red[t] += red[t + off];
    __syncthreads();
  }
  if (t == 0)
    out[(size_t)N_TOK * DIM] = (1.0f + BETA) * red[0] / (float)((size_t)N_TOK * DIM);
}

extern "C" void kernel_launch(void* const* d_in, const int* in_sizes, int n_in,
                              void* d_out, int out_size, void* d_ws, size_t ws_size,
                              hipStream_t stream) {
  const float* x = (const float*)d_in[0];
  const float* w = (const float*)d_in[1];
  float* out = (float*)d_out;
  float* ws  = (float*)d_ws;

  k_init<<<1, 256, 0, stream>>>(ws);
  k_minmax<<<NBLK, 256, 0, stream>>>(x, w, ws);
  k_setup<<<1, 1, 0, stream>>>(ws);
  for (int it = 0; it < 3; ++it) {
    k_sink<<<NBLK, 256, 0, stream>>>(x, w, ws);
    k_colreduce<<<KC, 256, 0, stream>>>(ws);
  }
  k_final<<<NBLK, 256, 0, stream>>>(x, w, out, ws);
  k_loss<<<1, 256, 0, stream>>>(out, ws);
}